// PatchTransformer_15487652069869
// MI455X (gfx1250) — compile-verified
//
#include <hip/hip_runtime.h>
#include <hip/hip_bf16.h>
#include <stdint.h>

// Problem constants (from reference setup_inputs): B=16, H=W=512, N=48,
// patch 1x3x256x256, RATIO_H = RATIO_W = 0.23.
#define HW    512
#define PW0   256
#define RATIO 0.23f

// Per-target parameter record: 16 floats (64 B, 4 x b128 async loads).
// [0]=img id, [1]=valid, [2]=cx, [3]=cy, [4]=cos, [5]=sin,
// [6]=0.5*pw, [7]=0.5*ph, [8]=256/max(pw,1), [9]=256/max(ph,1),
// [10..13] = bbox x0,x1,y0,y1, [14..15] pad.

__global__ void PatchTransformer_prep_kernel(const float* __restrict__ targets,
                                             const float* __restrict__ angles,
                                             float* __restrict__ params, int N) {
    int n = threadIdx.x;
    if (n >= N) return;
    const float* t = targets + n * 6;
    float idf = t[0];
    float xc  = floorf(t[2] * (float)HW);
    float yc  = floorf(t[3] * (float)HW);
    float bh  = t[5] * (float)HW;
    float ph  = floorf(bh * RATIO);
    float pw  = floorf(bh * RATIO);
    float rad = angles[n] * 0.017453292519943295f;  // |rad| <= 0.1745 -> fast HW path exact
    float sa, ca;
    __sincosf(rad, &sa, &ca);                       // v_sin_f32 / v_cos_f32, no Payne-Hanek blob
    float aca = fabsf(ca), asa = fabsf(sa);
    float ew  = ceilf(pw * aca + ph * asa);
    float eh  = ceilf(pw * asa + ph * aca);
    // NOTE: reference intentionally mixes eh/ew here (left uses eh, top uses 0.6*ew)
    float left = xc - floorf(0.5f * eh);
    float top  = yc - floorf(0.6f * ew);
    float cx = left + 0.5f * ew;
    float cy = top  + 0.5f * eh;
    float valid = (ph < 1.0f || pw < 1.0f) ? 0.0f : 1.0f;
    float sx = (float)PW0 / fmaxf(pw, 1.0f);
    float sy = (float)PW0 / fmaxf(ph, 1.0f);
    const float m = 2.0f;  // bilinear + rounding margin
    float x0 = cx - 0.5f * ew - m, x1 = cx + 0.5f * ew + m;
    float y0 = cy - 0.5f * eh - m, y1 = cy + 0.5f * eh + m;
    if (valid == 0.0f) { x0 = 1e30f; x1 = -1e30f; y0 = 1e30f; y1 = -1e30f; }
    float* p = params + n * 16;
    p[0] = idf; p[1] = valid; p[2] = cx; p[3] = cy;
    p[4] = ca;  p[5] = sa;    p[6] = 0.5f * pw; p[7] = 0.5f * ph;
    p[8] = sx;  p[9] = sy;    p[10] = x0; p[11] = x1;
    p[12] = y0; p[13] = y1;   p[14] = 0.f; p[15] = 0.f;
}

__global__ void __launch_bounds__(256)
PatchTransformer_15487652069869_kernel(const float* __restrict__ patch,
                                       const float* __restrict__ params,
                                       float* __restrict__ out,
                                       int N, int maskOff) {
    __shared__ float sP[64 * 16];   // up to 64 targets' params
    __shared__ int   sHits[64];
    __shared__ int   sNHit;

    const int tx  = threadIdx.x;           // 0..31
    const int ty  = threadIdx.y;           // 0..7
    const int tid = ty * 32 + tx;          // 0..255
    const int b   = blockIdx.z;
    const int xbase = blockIdx.x * 128 + tx * 4;  // 4 consecutive pixels / thread
    const int y     = blockIdx.y * 8 + ty;

    // --- Stage the parameter table into LDS via CDNA5 async global->LDS ---
    const int nvec = N * 4;  // N * 64B / 16B
    if (tid < nvec) {
        uint32_t ldsOff = (uint32_t)(uintptr_t)(void*)sP + (uint32_t)(tid * 16);
        int gOff = tid * 16;
        asm volatile("global_load_async_to_lds_b128 %0, %1, %2"
                     :: "v"(ldsOff), "v"(gOff), "s"(params)
                     : "memory");
    }
    asm volatile("s_wait_asynccnt 0" ::: "memory");
    __syncthreads();

    // --- Deterministic per-tile culling (single thread => replay-stable order) ---
    if (tid == 0) {
        const float tx0 = (float)(blockIdx.x * 128), tx1 = tx0 + 127.0f;
        const float ty0 = (float)(blockIdx.y * 8),   ty1 = ty0 + 7.0f;
        int cnt = 0;
        for (int n = 0; n < N; ++n) {
            const float* p = sP + n * 16;
            if ((int)p[0] == b && p[1] != 0.0f &&
                p[11] >= tx0 && p[10] <= tx1 &&
                p[13] >= ty0 && p[12] <= ty1) {
                sHits[cnt++] = n;
            }
        }
        sNHit = cnt;
    }
    __syncthreads();

    float acc0[4] = {0.f, 0.f, 0.f, 0.f};
    float acc1[4] = {0.f, 0.f, 0.f, 0.f};
    float acc2[4] = {0.f, 0.f, 0.f, 0.f};
    float cov[4]  = {0.f, 0.f, 0.f, 0.f};

    const int nh = sNHit;
    const float yf = (float)y;
    for (int k = 0; k < nh; ++k) {
        const float* p = sP + sHits[k] * 16;
        const float cx = p[2], cy = p[3], ca = p[4], sa = p[5];
        const float hpw = p[6], hph = p[7], sx = p[8], sy = p[9];
        const float dy = yf - cy;
#pragma unroll
        for (int j = 0; j < 4; ++j) {
            const float dx = (float)(xbase + j) - cx;
            const float xs = ca * dx + sa * dy + hpw;
            const float ys = ca * dy - sa * dx + hph;
            const float xp = xs * sx;
            const float yp = ys * sy;
            if (xp > -1.0f && xp < 256.0f && yp > -1.0f && yp < 256.0f) {
                const float xfl = floorf(xp), yfl = floorf(yp);
                const float fx = xp - xfl, fy = yp - yfl;
                const int x0i = (int)xfl, y0i = (int)yfl;
                const float w00 = (1.f - fx) * (1.f - fy);
                const float w01 = fx * (1.f - fy);
                const float w10 = (1.f - fx) * fy;
                const float w11 = fx * fy;
#pragma unroll
                for (int t = 0; t < 4; ++t) {
                    const int ddx = t & 1, ddy = t >> 1;
                    const int xi = x0i + ddx, yi = y0i + ddy;
                    const float wt = (t == 0) ? w00 : (t == 1) ? w01
                                   : (t == 2) ? w10 : w11;
                    if (xi >= 0 && xi < 256 && yi >= 0 && yi < 256) {
                        const int base = yi * 256 + xi;
                        acc0[j] += patch[base] * wt;
                        acc1[j] += patch[65536 + base] * wt;
                        acc2[j] += patch[131072 + base] * wt;
                        cov[j]  += wt;
                    }
                }
            }
        }
    }

    // --- Streaming float4 stores: patch_tmp (3 planes) + mask_tmp (3 planes) ---
    const int plane   = HW * HW;
    const int pixBase = y * HW + xbase;
    float4 o;
    o.x = fminf(fmaxf(cov[0], 0.f), 1.f);
    o.y = fminf(fmaxf(cov[1], 0.f), 1.f);
    o.z = fminf(fmaxf(cov[2], 0.f), 1.f);
    o.w = fminf(fmaxf(cov[3], 0.f), 1.f);
    float* mbase = out + maskOff + (b * 3) * plane + pixBase;
    *(float4*)(mbase)             = o;
    *(float4*)(mbase + plane)     = o;
    *(float4*)(mbase + 2 * plane) = o;

    float* pbase = out + (b * 3) * plane + pixBase;
    o.x = fminf(fmaxf(acc0[0], 0.f), 1.f);
    o.y = fminf(fmaxf(acc0[1], 0.f), 1.f);
    o.z = fminf(fmaxf(acc0[2], 0.f), 1.f);
    o.w = fminf(fmaxf(acc0[3], 0.f), 1.f);
    *(float4*)(pbase) = o;
    o.x = fminf(fmaxf(acc1[0], 0.f), 1.f);
    o.y = fminf(fmaxf(acc1[1], 0.f), 1.f);
    o.z = fminf(fmaxf(acc1[2], 0.f), 1.f);
    o.w = fminf(fmaxf(acc1[3], 0.f), 1.f);
    *(float4*)(pbase + plane) = o;
    o.x = fminf(fmaxf(acc2[0], 0.f), 1.f);
    o.y = fminf(fmaxf(acc2[1], 0.f), 1.f);
    o.z = fminf(fmaxf(acc2[2], 0.f), 1.f);
    o.w = fminf(fmaxf(acc2[3], 0.f), 1.f);
    *(float4*)(pbase + 2 * plane) = o;
}

extern "C" void kernel_launch(void* const* d_in, const int* in_sizes, int n_in,
                              void* d_out, int out_size, void* d_ws, size_t ws_size,
                              hipStream_t stream) {
    // setup_inputs order: patch[1,3,256,256], targets[N,6], imgs[B,3,512,512], angles[N]
    const float* patch   = (const float*)d_in[0];
    const float* targets = (const float*)d_in[1];
    const float* angles  = (const float*)d_in[3];
    const int N = in_sizes[1] / 6;                 // 48
    const int B = in_sizes[2] / (3 * HW * HW);     // 16

    float* params = (float*)d_ws;                  // N*16 floats (3 KB)

    PatchTransformer_prep_kernel<<<1, 256, 0, stream>>>(targets, angles, params, N);

    dim3 grid(HW / 128, HW / 8, B);   // (4, 64, 16) blocks
    dim3 block(32, 8, 1);             // 256 threads = 8 wave32
    const int maskOff = B * 3 * HW * HW;
    PatchTransformer_15487652069869_kernel<<<grid, block, 0, stream>>>(
        patch, params, (float*)d_out, N, maskOff);
}